// MolGNN_34316788695884
// MI455X (gfx1250) — compile-verified
//
#include <hip/hip_runtime.h>

// ---------------------------------------------------------------------------
// MolGNN on gfx1250: dense GEMMs via V_WMMA_F32_16X16X4_F32 (fp32 exact),
// wave32 tiling with M-register-blocking (one B fragment feeds 2-4 WMMAs),
// LDS row padding (+2) for conflict-free A-fragment ds_load_b64,
// edge scatter via hardware fp32 global atomics.
// ---------------------------------------------------------------------------

#define NN 200000
#define EE 800000
#define GG 8192
#define HH 128
#define OO 256
#define LL 3

typedef float v2f __attribute__((ext_vector_type(2)));
typedef float v8f __attribute__((ext_vector_type(8)));

__device__ __forceinline__ v8f wmma4(v2f a, v2f b, v8f c) {
  // 8-arg pattern: (neg_a, A, neg_b, B, c_mod, C, reuse_a, reuse_b)
  return __builtin_amdgcn_wmma_f32_16x16x4_f32(false, a, false, b, (short)0, c,
                                               false, false);
}

// MB stacked 16x16 output tiles per wave, sharing each B fragment.
// As: LDS [MB*16 rows, K cols] with row stride lda (pad => conflict-free).
// W : global row-major [K, ldw].
template <int MB>
__device__ __forceinline__ void tile_gemm_mb(const float* As, int lda, int K,
                                             const float* __restrict__ W,
                                             int ldw, int col0, int r, int half,
                                             v8f* acc) {
  for (int kk = 0; kk < K; kk += 4) {
    v2f b;
    // B 4x16 fp32 fragment: rows kk+2*half / +1 striped across lanes (col = r)
    b.x = W[(size_t)(kk + 2 * half) * ldw + col0 + r];
    b.y = W[(size_t)(kk + 2 * half + 1) * ldw + col0 + r];
#pragma unroll
    for (int m = 0; m < MB; ++m) {
      v2f a;
      // A 16x4 fp32 fragment: VGPR0 = K+2*half, VGPR1 = +1 (ISA 7.12.2)
      a.x = As[(m * 16 + r) * lda + kk + 2 * half];
      a.y = As[(m * 16 + r) * lda + kk + 2 * half + 1];
      acc[m] = wmma4(a, b, acc[m]);
    }
  }
}

// ---- node embedding gather + [32,288]x[288,128] GEMM + bias (MB=2) ---------
__global__ __launch_bounds__(256) void node_embed_kernel(
    const int* __restrict__ x, const float* __restrict__ e0,
    const float* __restrict__ e1, const float* __restrict__ e2,
    const float* __restrict__ e3, const float* __restrict__ e4,
    const float* __restrict__ e5, const float* __restrict__ e6,
    const float* __restrict__ e7, const float* __restrict__ e8,
    const float* __restrict__ Wn, const float* __restrict__ bn,
    float* __restrict__ h) {
  const int LDA = 288 + 2;
  __shared__ float As[32 * (288 + 2)];
  const int row0 = blockIdx.x * 32;
  const int tid = threadIdx.x;
  const float* tabs[9] = {e0, e1, e2, e3, e4, e5, e6, e7, e8};
  for (int i = tid; i < 32 * 288; i += 256) {
    int nloc = i / 288, kp = i - nloc * 288;
    int f = kp >> 5, d = kp & 31;
    int idx = x[(size_t)(row0 + nloc) * 9 + f];
    As[nloc * LDA + kp] = tabs[f][(size_t)idx * 32 + d];
  }
  __syncthreads();
  const int wave = tid >> 5, lane = tid & 31, r = lane & 15, half = lane >> 4;
  const int col0 = wave * 16;
  v8f acc[2] = {{0.f, 0.f, 0.f, 0.f, 0.f, 0.f, 0.f, 0.f},
                {0.f, 0.f, 0.f, 0.f, 0.f, 0.f, 0.f, 0.f}};
  tile_gemm_mb<2>(As, LDA, 288, Wn, HH, col0, r, half, acc);
  const float bb = bn[col0 + r];
#pragma unroll
  for (int m = 0; m < 2; ++m)
#pragma unroll
    for (int j = 0; j < 8; ++j) {
      int row = row0 + m * 16 + j + 8 * half;
      h[(size_t)row * HH + col0 + r] = acc[m][j] + bb;
    }
}

// ---- edge embedding gather + [64,96]x[96,128] GEMM + bias (MB=4) -----------
__global__ __launch_bounds__(256) void edge_embed_kernel(
    const int* __restrict__ ea, const float* __restrict__ t0,
    const float* __restrict__ t1, const float* __restrict__ t2,
    const float* __restrict__ We, const float* __restrict__ be,
    float* __restrict__ e) {
  const int LDA = 96 + 2;
  __shared__ float As[64 * (96 + 2)];
  const int row0 = blockIdx.x * 64;
  const int tid = threadIdx.x;
  const float* tabs[3] = {t0, t1, t2};
  for (int i = tid; i < 64 * 96; i += 256) {
    int eloc = i / 96, kp = i - eloc * 96;
    int f = kp >> 5, d = kp & 31;
    int idx = ea[(size_t)(row0 + eloc) * 3 + f];
    As[eloc * LDA + kp] = tabs[f][(size_t)idx * 32 + d];
  }
  __syncthreads();
  const int wave = tid >> 5, lane = tid & 31, r = lane & 15, half = lane >> 4;
  const int col0 = wave * 16;
  v8f acc[4] = {{0.f, 0.f, 0.f, 0.f, 0.f, 0.f, 0.f, 0.f},
                {0.f, 0.f, 0.f, 0.f, 0.f, 0.f, 0.f, 0.f},
                {0.f, 0.f, 0.f, 0.f, 0.f, 0.f, 0.f, 0.f},
                {0.f, 0.f, 0.f, 0.f, 0.f, 0.f, 0.f, 0.f}};
  tile_gemm_mb<4>(As, LDA, 96, We, HH, col0, r, half, acc);
  const float bb = be[col0 + r];
#pragma unroll
  for (int m = 0; m < 4; ++m)
#pragma unroll
    for (int j = 0; j < 8; ++j) {
      int row = row0 + m * 16 + j + 8 * half;
      e[(size_t)row * HH + col0 + r] = acc[m][j] + bb;
    }
}

// ---- zero fill (float4) ----------------------------------------------------
__global__ void fill_zero_kernel(float4* __restrict__ p, long n4) {
  long i = (long)blockIdx.x * blockDim.x + threadIdx.x;
  if (i < n4) p[i] = make_float4(0.f, 0.f, 0.f, 0.f);
}

// ---- msg = relu(h[src] + e), agg[dst] += msg (fp32 hw atomics) -------------
__global__ void msg_scatter_kernel(const float* __restrict__ h,
                                   const float* __restrict__ e,
                                   const int* __restrict__ src,
                                   const int* __restrict__ dst,
                                   float* __restrict__ agg, int nE) {
  long gid = (long)blockIdx.x * blockDim.x + threadIdx.x;
  int ei = (int)(gid >> 5);
  if (ei >= nE) return;
  int c4 = (int)(gid & 31) * 4;
  int s = src[ei], d = dst[ei];
  float4 hv = *(const float4*)(h + (size_t)s * HH + c4);
  float4 ev = *(const float4*)(e + (size_t)ei * HH + c4);
  float4 m;
  m.x = fmaxf(hv.x + ev.x, 0.f);
  m.y = fmaxf(hv.y + ev.y, 0.f);
  m.z = fmaxf(hv.z + ev.z, 0.f);
  m.w = fmaxf(hv.w + ev.w, 0.f);
  float* ap = agg + (size_t)d * HH + c4;
  unsafeAtomicAdd(ap + 0, m.x);
  unsafeAtomicAdd(ap + 1, m.y);
  unsafeAtomicAdd(ap + 2, m.z);
  unsafeAtomicAdd(ap + 3, m.w);
}

// ---- fused GINE MLP: h_out = relu(relu((h+agg)@W1+b1)@W2+b2), MB=4 ---------
// Single 64x(128+2) LDS buffer reused for both GEMM inputs (barrier-separated:
// all of As is consumed into accumulators before z1 is written back).
__global__ __launch_bounds__(256) void conv_mlp_kernel(
    const float* __restrict__ h, const float* __restrict__ agg,
    const float* __restrict__ W1, const float* __restrict__ b1,
    const float* __restrict__ W2, const float* __restrict__ b2,
    float* __restrict__ hout) {
  const int LDA = HH + 2;
  __shared__ float As[64 * (HH + 2)];
  const int row0 = blockIdx.x * 64;
  const int tid = threadIdx.x;
  for (int i = tid; i < 64 * HH; i += 256) {
    size_t gi = (size_t)row0 * HH + i;
    As[(i >> 7) * LDA + (i & 127)] = h[gi] + agg[gi];
  }
  __syncthreads();
  const int wave = tid >> 5, lane = tid & 31, r = lane & 15, half = lane >> 4;
  const int col0 = wave * 16;
  v8f acc[4] = {{0.f, 0.f, 0.f, 0.f, 0.f, 0.f, 0.f, 0.f},
                {0.f, 0.f, 0.f, 0.f, 0.f, 0.f, 0.f, 0.f},
                {0.f, 0.f, 0.f, 0.f, 0.f, 0.f, 0.f, 0.f},
                {0.f, 0.f, 0.f, 0.f, 0.f, 0.f, 0.f, 0.f}};
  // z1 = relu((h+agg)@W1 + b1)
  tile_gemm_mb<4>(As, LDA, HH, W1, HH, col0, r, half, acc);
  __syncthreads();  // everyone done reading As before overwrite
  {
    const float bb = b1[col0 + r];
#pragma unroll
    for (int m = 0; m < 4; ++m)
#pragma unroll
      for (int j = 0; j < 8; ++j)
        As[(m * 16 + j + 8 * half) * LDA + col0 + r] =
            fmaxf(acc[m][j] + bb, 0.f);
  }
  __syncthreads();
  // h_out = relu(z1@W2 + b2)
#pragma unroll
  for (int m = 0; m < 4; ++m) acc[m] = {0.f, 0.f, 0.f, 0.f, 0.f, 0.f, 0.f, 0.f};
  tile_gemm_mb<4>(As, LDA, HH, W2, HH, col0, r, half, acc);
  {
    const float bb = b2[col0 + r];
#pragma unroll
    for (int m = 0; m < 4; ++m)
#pragma unroll
      for (int j = 0; j < 8; ++j) {
        int row = row0 + m * 16 + j + 8 * half;
        hout[(size_t)row * HH + col0 + r] = fmaxf(acc[m][j] + bb, 0.f);
      }
  }
}

// ---- global add pool: g[batch[n]] += h[n] ----------------------------------
__global__ void pool_kernel(const float* __restrict__ h,
                            const int* __restrict__ batch,
                            float* __restrict__ g, int nN) {
  long gid = (long)blockIdx.x * blockDim.x + threadIdx.x;
  int n = (int)(gid >> 5);
  if (n >= nN) return;
  int c4 = (int)(gid & 31) * 4;
  int b = batch[n];
  float4 hv = *(const float4*)(h + (size_t)n * HH + c4);
  float* gp = g + (size_t)b * HH + c4;
  unsafeAtomicAdd(gp + 0, hv.x);
  unsafeAtomicAdd(gp + 1, hv.y);
  unsafeAtomicAdd(gp + 2, hv.z);
  unsafeAtomicAdd(gp + 3, hv.w);
}

// ---- out = normalize(g @ Wp + bp) : [16,128]x[128,256] + row L2 norm -------
__global__ __launch_bounds__(256) void proj_norm_kernel(
    const float* __restrict__ g, const float* __restrict__ Wp,
    const float* __restrict__ bp, float* __restrict__ out) {
  const int LDA = HH + 2;
  __shared__ float As[16 * (HH + 2)];
  __shared__ float Os[16 * OO];
  __shared__ float nrm[16];
  const int row0 = blockIdx.x * 16;
  const int tid = threadIdx.x;
  for (int i = tid; i < 16 * HH; i += 256)
    As[(i >> 7) * LDA + (i & 127)] = g[(size_t)row0 * HH + i];
  __syncthreads();
  const int wave = tid >> 5, lane = tid & 31, r = lane & 15, half = lane >> 4;
#pragma unroll
  for (int t = 0; t < 2; ++t) {
    const int col0 = wave * 16 + t * 128;
    v8f acc[1] = {{0.f, 0.f, 0.f, 0.f, 0.f, 0.f, 0.f, 0.f}};
    tile_gemm_mb<1>(As, LDA, HH, Wp, OO, col0, r, half, acc);
    const float bb = bp[col0 + r];
#pragma unroll
    for (int j = 0; j < 8; ++j)
      Os[(j + 8 * half) * OO + col0 + r] = acc[0][j] + bb;
  }
  __syncthreads();
  if (tid < 16) {
    float s = 0.f;
    for (int c = 0; c < OO; ++c) {
      float v = Os[tid * OO + c];
      s += v * v;
    }
    nrm[tid] = fmaxf(sqrtf(s), 1e-12f);
  }
  __syncthreads();
  for (int i = tid; i < 16 * OO; i += 256) {
    int rr = i >> 8;  // / 256
    out[(size_t)(row0 + rr) * OO + (i & 255)] = Os[i] / nrm[rr];
  }
}

// ---------------------------------------------------------------------------
extern "C" void kernel_launch(void* const* d_in, const int* in_sizes, int n_in,
                              void* d_out, int out_size, void* d_ws, size_t ws_size,
                              hipStream_t stream) {
  const int* x          = (const int*)d_in[0];
  const int* edge_attr  = (const int*)d_in[1];
  const int* edge_index = (const int*)d_in[2];
  const int* batch_ids  = (const int*)d_in[3];
  const float* emb[9];
  for (int i = 0; i < 9; ++i) emb[i] = (const float*)d_in[4 + i];
  const float* node_w  = (const float*)d_in[13];
  const float* node_b  = (const float*)d_in[14];
  const float* ebond   = (const float*)d_in[15];
  const float* estereo = (const float*)d_in[16];
  const float* econj   = (const float*)d_in[17];
  const float* edge_w  = (const float*)d_in[18];
  const float* edge_b  = (const float*)d_in[19];
  const float* conv_w1 = (const float*)d_in[20];
  const float* conv_b1 = (const float*)d_in[21];
  const float* conv_w2 = (const float*)d_in[22];
  const float* conv_b2 = (const float*)d_in[23];
  const float* proj_w  = (const float*)d_in[24];
  const float* proj_b  = (const float*)d_in[25];
  float* out = (float*)d_out;

  // workspace partition
  char* ws = (char*)d_ws;
  size_t off = 0;
  auto walloc = [&](size_t bytes) -> float* {
    off = (off + 255) & ~(size_t)255;
    char* p = ws + off;
    off += bytes;
    return (float*)p;
  };
  float* h_a  = walloc((size_t)NN * HH * 4);
  float* h_b  = walloc((size_t)NN * HH * 4);
  float* agg  = walloc((size_t)NN * HH * 4);
  float* eemb = walloc((size_t)EE * HH * 4);
  float* gbuf = walloc((size_t)GG * HH * 4);

  node_embed_kernel<<<NN / 32, 256, 0, stream>>>(
      x, emb[0], emb[1], emb[2], emb[3], emb[4], emb[5], emb[6], emb[7], emb[8],
      node_w, node_b, h_a);
  edge_embed_kernel<<<EE / 64, 256, 0, stream>>>(
      edge_attr, ebond, estereo, econj, edge_w, edge_b, eemb);

  const int* src = edge_index;
  const int* dst = edge_index + EE;
  float* hc = h_a;
  float* hn = h_b;
  for (int l = 0; l < LL; ++l) {
    long n4 = (long)NN * HH / 4;
    fill_zero_kernel<<<(n4 + 255) / 256, 256, 0, stream>>>((float4*)agg, n4);
    long mt = (long)EE * 32;
    msg_scatter_kernel<<<(mt + 255) / 256, 256, 0, stream>>>(hc, eemb, src, dst,
                                                             agg, EE);
    conv_mlp_kernel<<<NN / 64, 256, 0, stream>>>(
        hc, agg, conv_w1 + (size_t)l * HH * HH, conv_b1 + (size_t)l * HH,
        conv_w2 + (size_t)l * HH * HH, conv_b2 + (size_t)l * HH, hn);
    float* t = hc; hc = hn; hn = t;
  }

  long g4 = (long)GG * HH / 4;
  fill_zero_kernel<<<(g4 + 255) / 256, 256, 0, stream>>>((float4*)gbuf, g4);
  long pt = (long)NN * 32;
  pool_kernel<<<(pt + 255) / 256, 256, 0, stream>>>(hc, batch_ids, gbuf, NN);
  proj_norm_kernel<<<GG / 16, 256, 0, stream>>>(gbuf, proj_w, proj_b, out);
}